// FasterRCNNBase_50568944943273
// MI455X (gfx1250) — compile-verified
//
#include <hip/hip_runtime.h>
#include <cstdint>

#define RR 1000          // real RoIs
#define RP 1024          // padded to power of two for bitonic sort
#define NC 81            // classes incl. background
#define NFG 80           // foreground classes
#define IMGH 600.0f
#define IMGW 800.0f
#define STH 0.05f        // score threshold
#define NTH 0.5f         // NMS IoU threshold

// ---------- CDNA5 async global->LDS helpers (gfx1250 VGLOBAL async ops) ----------
// GVS mode: mem_addr = SGPR64 + VGPR32 (+ inst offset); first operand is the LDS
// byte-address VGPR (VDST), tracked with ASYNCcnt.
__device__ __forceinline__ uint32_t lds_off(const void* p) {
    // generic pointer to LDS: low 32 bits are the LDS byte offset (ISA 10.2 aperture rules)
    return (uint32_t)(uint64_t)(uintptr_t)p;
}
__device__ __forceinline__ void async_ld_b32(uint32_t ldsOff, uint32_t byteOff, const void* base) {
    asm volatile("global_load_async_to_lds_b32 %0, %1, %2"
                 :: "v"(ldsOff), "v"(byteOff), "s"(base) : "memory");
}
__device__ __forceinline__ void async_ld_b128(uint32_t ldsOff, uint32_t byteOff, const void* base) {
    asm volatile("global_load_async_to_lds_b128 %0, %1, %2"
                 :: "v"(ldsOff), "v"(byteOff), "s"(base) : "memory");
}
__device__ __forceinline__ void wait_async0() {
    asm volatile("s_wait_asynccnt 0" ::: "memory");
}

// ---------------- Kernel 1: box decode + clamp ----------------
// One block per RoI (blockIdx = r), one lane per class (no integer division).
__global__ __launch_bounds__(128) void frcnn_decode(const float* __restrict__ roi,
                                                    const float* __restrict__ roi_loc,
                                                    float* __restrict__ bbox) {
    int r = blockIdx.x;
    int c = threadIdx.x;
    if (c >= NC) return;

    const float4 rb = *(const float4*)(roi + (size_t)r * 4);  // uniform per block
    float h = rb.z - rb.x, w = rb.w - rb.y;
    float cy = rb.x + 0.5f * h, cx = rb.y + 0.5f * w;

    const float4 l = *(const float4*)(roi_loc + (size_t)r * NC * 4 + (size_t)c * 4);
    float dy = l.x * 0.1f, dx = l.y * 0.1f;
    float dh = l.z * 0.2f, dw = l.w * 0.2f;

    float ny = dy * h + cy;
    float nx = dx * w + cx;
    float nh = __expf(dh) * h;
    float nw = __expf(dw) * w;

    float4 b;
    b.x = fminf(fmaxf(ny - 0.5f * nh, 0.f), IMGH);
    b.y = fminf(fmaxf(nx - 0.5f * nw, 0.f), IMGW);
    b.z = fminf(fmaxf(ny + 0.5f * nh, 0.f), IMGH);
    b.w = fminf(fmaxf(nx + 0.5f * nw, 0.f), IMGW);
    *(float4*)(bbox + ((size_t)r * NC + c) * 4) = b;
}

// ---------------- Kernel 2: softmax over 81 classes, one wave per RoI ----------------
__global__ __launch_bounds__(256) void frcnn_softmax(const float* __restrict__ score,
                                                     float* __restrict__ prob) {
    int lane = threadIdx.x & 31;
    int wave = threadIdx.x >> 5;
    int r = blockIdx.x * 8 + wave;
    if (r >= RR) return;
    const float* s = score + (size_t)r * NC;

    float v0 = s[lane];
    float v1 = s[lane + 32];
    float v2 = (lane + 64 < NC) ? s[lane + 64] : -__builtin_huge_valf();

    float m = fmaxf(fmaxf(v0, v1), v2);
    for (int o = 16; o; o >>= 1) m = fmaxf(m, __shfl_xor(m, o, 32));

    float e0 = __expf(v0 - m);
    float e1 = __expf(v1 - m);
    float e2 = (lane + 64 < NC) ? __expf(v2 - m) : 0.f;
    float sum = e0 + e1 + e2;
    for (int o = 16; o; o >>= 1) sum += __shfl_xor(sum, o, 32);
    float inv = 1.0f / sum;

    prob[(size_t)r * NC + lane]      = e0 * inv;
    prob[(size_t)r * NC + lane + 32] = e1 * inv;
    if (lane + 64 < NC) prob[(size_t)r * NC + lane + 64] = e2 * inv;
}

// ---------------- Kernel 3: per-class NMS, one workgroup per class ----------------
// Uses CDNA5's 320KB LDS: full 1024x1024-bit suppression matrix lives in LDS.
__global__ __launch_bounds__(1024) void frcnn_nms(const float* __restrict__ bbox,
                                                  const float* __restrict__ prob,
                                                  float* __restrict__ out,
                                                  int* __restrict__ labels) {
    extern __shared__ uint32_t s_mask[];          // RP * 32 words = 128 KB (dynamic LDS)
    __shared__ float               s_score[RP];
    __shared__ int                 s_idx[RP];
    __shared__ __align__(16) float s_box[RP][4];  // 16B rows for async B128 dest
    __shared__ float               s_area[RP];
    __shared__ uint32_t            s_keep[32];

    const int tid  = threadIdx.x;
    const int lane = tid & 31;
    const int wv   = tid >> 5;
    const int cls  = blockIdx.x + 1;   // skip background class 0

    // ---- stage this class's scores into LDS (async, strided gather) ----
    if (tid < RR) {
        async_ld_b32(lds_off(&s_score[tid]), (uint32_t)((tid * NC + cls) * 4), prob);
    } else {
        s_score[tid] = -__builtin_huge_valf();   // sort padding to the tail
    }
    s_idx[tid] = tid;
    wait_async0();
    __syncthreads();

    // ---- bitonic sort 1024 elems, descending by score, index payload ----
    for (int k = 2; k <= RP; k <<= 1) {
        for (int j = k >> 1; j > 0; j >>= 1) {
            if (tid < RP / 2) {
                int i   = ((tid & ~(j - 1)) << 1) | (tid & (j - 1));
                int ixj = i | j;
                bool desc = (i & k) == 0;
                float a = s_score[i], b = s_score[ixj];
                if ((a < b) == desc) {
                    s_score[i] = b; s_score[ixj] = a;
                    int t = s_idx[i]; s_idx[i] = s_idx[ixj]; s_idx[ixj] = t;
                }
            }
            __syncthreads();
        }
    }

    // ---- gather sorted boxes into LDS (async B128 per lane) ----
    int orig = s_idx[tid];
    if (orig < RR) {
        async_ld_b128(lds_off(&s_box[tid][0]), (uint32_t)((orig * NC + cls) * 16), bbox);
    } else {
        s_box[tid][0] = s_box[tid][1] = s_box[tid][2] = s_box[tid][3] = 0.f;
    }
    wait_async0();
    __syncthreads();

    s_area[tid] = (s_box[tid][2] - s_box[tid][0]) * (s_box[tid][3] - s_box[tid][1]);

    // initial keep = score > threshold (wave32 ballot -> one word per wave)
    uint32_t vb = __builtin_amdgcn_ballot_w32(s_score[tid] > STH);
    if (lane == 0) s_keep[wv] = vb;
    __syncthreads();

    // ---- build suppression bitmask: mask[row][w] bit l = (IoU(row, w*32+l) > thr && col > row) ----
    for (int row = wv; row < RP; row += 32) {
        float ry0 = s_box[row][0], rx0 = s_box[row][1];
        float ry1 = s_box[row][2], rx1 = s_box[row][3];
        float rar = s_area[row];
        for (int w = 0; w < 32; ++w) {
            int col = (w << 5) | lane;
            float ty = fmaxf(ry0, s_box[col][0]);
            float tx = fmaxf(rx0, s_box[col][1]);
            float by = fminf(ry1, s_box[col][2]);
            float bx = fminf(rx1, s_box[col][3]);
            float ih = fmaxf(by - ty, 0.f);
            float iw = fmaxf(bx - tx, 0.f);
            float inter = ih * iw;
            float iou = inter / (rar + s_area[col] - inter + 1e-9f);
            uint32_t m = __builtin_amdgcn_ballot_w32((iou > NTH) && (col > row));
            if (lane == 0) s_mask[row * 32 + w] = m;
        }
    }
    __syncthreads();

    // ---- serial greedy scan, executed by wave 0 (32 lanes own the 32 keep-words) ----
    if (tid < 32) {
        volatile uint32_t* vk = s_keep;
        for (int i = 0; i < RR; ++i) {
            uint32_t kw = vk[i >> 5];          // uniform read across the wave
            if (kw & (1u << (i & 31))) {
                vk[lane] &= ~s_mask[i * 32 + lane];
            }
        }
    }
    __syncthreads();

    // ---- emit masked detections + labels ----
    if (tid < RR) {
        bool kept = (s_keep[tid >> 5] >> (tid & 31)) & 1u;
        float f = kept ? 1.f : 0.f;
        int c = blockIdx.x;                      // label index 0..79
        float* o = out + ((size_t)c * RR + tid) * 5;
        o[0] = s_box[tid][0] * f;
        o[1] = s_box[tid][1] * f;
        o[2] = s_box[tid][2] * f;
        o[3] = s_box[tid][3] * f;
        o[4] = s_score[tid] * f;
        labels[(size_t)c * RR + tid] = kept ? c : -1;
    }
}

extern "C" void kernel_launch(void* const* d_in, const int* in_sizes, int n_in,
                              void* d_out, int out_size, void* d_ws, size_t ws_size,
                              hipStream_t stream) {
    (void)in_sizes; (void)n_in; (void)out_size; (void)ws_size;
    const float* roi       = (const float*)d_in[0];   // [1000,4]
    const float* roi_loc   = (const float*)d_in[1];   // [1000,324]
    const float* roi_score = (const float*)d_in[2];   // [1000,81]

    float* bbox = (float*)d_ws;                       // [1000,81,4]
    float* prob = bbox + (size_t)RR * NC * 4;         // [1000,81]

    float* out    = (float*)d_out;                     // [80,1000,5]
    int*   labels = (int*)(out + (size_t)NFG * RR * 5);// [80,1000] as int32 bits

    // allow >64KB dynamic LDS (CDNA5 WGP has 320KB); not a stream op, capture-safe
    hipFuncSetAttribute((const void*)frcnn_nms,
                        hipFuncAttributeMaxDynamicSharedMemorySize, RP * 32 * 4);

    frcnn_decode<<<RR, 128, 0, stream>>>(roi, roi_loc, bbox);
    frcnn_softmax<<<(RR + 7) / 8, 256, 0, stream>>>(roi_score, prob);
    frcnn_nms<<<NFG, 1024, RP * 32 * 4, stream>>>(bbox, prob, out, labels);
}